// GraphConvolutionalNetwork_77747497992412
// MI455X (gfx1250) — compile-verified
//
#include <hip/hip_runtime.h>

#define N_NODES 50000
#define N_EDGES 800000
#define GCN_EPS 1e-5f

typedef __attribute__((ext_vector_type(2))) float v2f;
typedef __attribute__((ext_vector_type(8))) float v8f;

// ---------------------------------------------------------------------------
// Edge preprocessing
// ---------------------------------------------------------------------------
__global__ void init_deg_kernel(float* deg) {
    int i = blockIdx.x * blockDim.x + threadIdx.x;
    if (i < N_NODES) deg[i] = 1.0f;  // self-loop contributes 1
}

__global__ void prep_edges_kernel(const long long* __restrict__ ei,
                                  int* __restrict__ src32, int* __restrict__ dst32,
                                  float* __restrict__ deg) {
    int e = blockIdx.x * blockDim.x + threadIdx.x;
    if (e < N_EDGES) {
        int s = (int)ei[e];
        int d = (int)ei[N_EDGES + e];
        src32[e] = s;
        dst32[e] = d;
        unsafeAtomicAdd(&deg[d], 1.0f);
    }
}

__global__ void dinv_kernel(float* deg_dinv) {
    int i = blockIdx.x * blockDim.x + threadIdx.x;
    if (i < N_NODES) {
        float dg = deg_dinv[i];
        deg_dinv[i] = (dg > 0.0f) ? rsqrtf(dg) : 0.0f;
    }
}

__global__ void norm_kernel(const int* __restrict__ src32, const int* __restrict__ dst32,
                            const float* __restrict__ dinv, float* __restrict__ norm) {
    int e = blockIdx.x * blockDim.x + threadIdx.x;
    if (e < N_EDGES) norm[e] = dinv[src32[e]] * dinv[dst32[e]];
}

// ---------------------------------------------------------------------------
// WMMA fp32 GEMM with fused self-loop epilogue:
//   xw[N x fout]  = A[N x fin] @ W[fin x fout]
//   agg[N x fout] = xw * dinv[row]^2        (aggregation buffer init)
// One wave per 16x16 output tile, K stepped by 4 (V_WMMA_F32_16X16X4_F32).
// ---------------------------------------------------------------------------
__global__ void gemm_wmma_f32(const float* __restrict__ A, const float* __restrict__ W,
                              float* __restrict__ xw, float* __restrict__ agg,
                              const float* __restrict__ dinv, int fin, int fout) {
    const int lane = threadIdx.x & 31;
    const int wave = threadIdx.x >> 5;
    const int mtile = blockIdx.x * (blockDim.x >> 5) + wave;   // wave-uniform
    const int M0 = mtile * 16;
    if (M0 >= N_NODES) return;                                  // uniform branch: EXEC stays full
    const int N0 = blockIdx.y * 16;

    const int half = lane >> 4;   // 0: K={0,1}; 1: K={2,3}
    const int l = lane & 15;

    v8f c = {};
    const float* arow = A + (size_t)(M0 + l) * fin + half * 2;  // 8B aligned (fin % 4 == 0)
    const float* bcol = W + (size_t)(half * 2) * fout + N0 + l;

    for (int k = 0; k < fin; k += 4) {
        v2f a = *(const v2f*)(arow + k);                        // A[M0+l][k+2*half .. +1]
        v2f b;
        b.x = bcol[(size_t)k * fout];                           // W[k+2*half  ][N0+l]
        b.y = bcol[(size_t)(k + 1) * fout];                     // W[k+2*half+1][N0+l]
        c = __builtin_amdgcn_wmma_f32_16x16x4_f32(
                /*neg_a=*/false, a, /*neg_b=*/false, b,
                /*c_mod=*/(short)0, c, /*reuse_a=*/false, /*reuse_b=*/false);
    }

    // C/D layout: VGPR v -> row M0+v (lanes 0-15) or M0+v+8 (lanes 16-31), col N0+l
    const int row0 = M0 + half * 8;
    float* ox = xw  + (size_t)row0 * fout + N0 + l;
    float* oa = agg + (size_t)row0 * fout + N0 + l;
#pragma unroll
    for (int v = 0; v < 8; ++v) {
        float d = dinv[row0 + v];          // uniform across lanes 0-15 / 16-31
        ox[(size_t)v * fout] = c[v];
        oa[(size_t)v * fout] = c[v] * d * d;
    }
}

// ---------------------------------------------------------------------------
// Edge scatter with HW f32 atomics (float4 chunks; F/4 threads per edge)
// ---------------------------------------------------------------------------
template <int F>   // F % 4 == 0
__global__ void scatter_vec4_kernel(const float* __restrict__ xw,
                                    const int* __restrict__ src, const int* __restrict__ dst,
                                    const float* __restrict__ norm, float* __restrict__ out) {
    constexpr int CH = F / 4;
    int t = blockIdx.x * blockDim.x + threadIdx.x;
    if (t >= N_EDGES * CH) return;
    int e = t / CH;          // power-of-2 -> shift
    int c = t - e * CH;
    int s = src[e], d = dst[e];
    float w = norm[e];
    float4 v = *((const float4*)(xw + (size_t)s * F) + c);
    float* op = out + (size_t)d * F + c * 4;
    unsafeAtomicAdd(op + 0, v.x * w);
    unsafeAtomicAdd(op + 1, v.y * w);
    unsafeAtomicAdd(op + 2, v.z * w);
    unsafeAtomicAdd(op + 3, v.w * w);
}

__global__ void scatter_f6_kernel(const float* __restrict__ xw,
                                  const int* __restrict__ src, const int* __restrict__ dst,
                                  const float* __restrict__ norm, float* __restrict__ out) {
    int e = blockIdx.x * blockDim.x + threadIdx.x;
    if (e >= N_EDGES) return;
    int s = src[e], d = dst[e];
    float w = norm[e];
    const float* xp = xw + (size_t)s * 6;
    float* op = out + (size_t)d * 6;
#pragma unroll
    for (int f = 0; f < 6; ++f) unsafeAtomicAdd(op + f, xp[f] * w);
}

// ---------------------------------------------------------------------------
// Fused bias + BatchNorm(eval) + ReLU
// ---------------------------------------------------------------------------
template <int F>
__global__ void bias_bn_relu_kernel(const float* __restrict__ in, float* __restrict__ out,
                                    const float* __restrict__ b, const float* __restrict__ g,
                                    const float* __restrict__ be, const float* __restrict__ m,
                                    const float* __restrict__ v) {
    int i = blockIdx.x * blockDim.x + threadIdx.x;
    if (i < N_NODES * F) {
        int f = i % F;       // power-of-2 -> mask
        float x = in[i] + b[f];
        x = (x - m[f]) * rsqrtf(v[f] + GCN_EPS) * g[f] + be[f];
        out[i] = fmaxf(x, 0.0f);
    }
}

// ---------------------------------------------------------------------------
// Layer 4 GEMM (64 -> 6): tiny, scalar with LDS-cached weights.
// Fused self-loop epilogue like the WMMA GEMM.
// ---------------------------------------------------------------------------
__global__ void gemm4_kernel(const float* __restrict__ h, const float* __restrict__ W,
                             float* __restrict__ xw, float* __restrict__ agg,
                             const float* __restrict__ dinv) {
    __shared__ float w[64 * 6];
    for (int i = threadIdx.x; i < 64 * 6; i += blockDim.x) w[i] = W[i];
    __syncthreads();
    int n = blockIdx.x * blockDim.x + threadIdx.x;
    if (n >= N_NODES) return;
    const float* hr = h + (size_t)n * 64;
    float acc[6] = {0.f, 0.f, 0.f, 0.f, 0.f, 0.f};
    for (int k = 0; k < 64; ++k) {
        float hv = hr[k];
#pragma unroll
        for (int j = 0; j < 6; ++j) acc[j] += hv * w[k * 6 + j];
    }
    float d = dinv[n];
    float d2 = d * d;
    float* ox = xw  + (size_t)n * 6;
    float* oa = agg + (size_t)n * 6;
#pragma unroll
    for (int j = 0; j < 6; ++j) {
        ox[j] = acc[j];
        oa[j] = acc[j] * d2;
    }
}

__global__ void final_bias_kernel(const float* __restrict__ agg, const float* __restrict__ b,
                                  float* __restrict__ out) {
    int i = blockIdx.x * blockDim.x + threadIdx.x;
    if (i < N_NODES * 6) out[i] = agg[i] + b[i % 6];
}

// ---------------------------------------------------------------------------
// Launch
// ---------------------------------------------------------------------------
static inline int cdiv_i(long long a, long long b) { return (int)((a + b - 1) / b); }

extern "C" void kernel_launch(void* const* d_in, const int* in_sizes, int n_in,
                              void* d_out, int out_size, void* d_ws, size_t ws_size,
                              hipStream_t stream) {
    const float*     x   = (const float*)d_in[0];
    const long long* ei  = (const long long*)d_in[1];
    const float* W1 = (const float*)d_in[2];
    const float* b1 = (const float*)d_in[3];
    const float* g1 = (const float*)d_in[4];
    const float* be1= (const float*)d_in[5];
    const float* m1 = (const float*)d_in[6];
    const float* v1 = (const float*)d_in[7];
    const float* W2 = (const float*)d_in[8];
    const float* b2 = (const float*)d_in[9];
    const float* g2 = (const float*)d_in[10];
    const float* be2= (const float*)d_in[11];
    const float* m2 = (const float*)d_in[12];
    const float* v2 = (const float*)d_in[13];
    const float* W3 = (const float*)d_in[14];
    const float* b3 = (const float*)d_in[15];
    const float* g3 = (const float*)d_in[16];
    const float* be3= (const float*)d_in[17];
    const float* m3 = (const float*)d_in[18];
    const float* v3 = (const float*)d_in[19];
    const float* W4 = (const float*)d_in[20];
    const float* b4 = (const float*)d_in[21];
    float* out = (float*)d_out;

    // Workspace layout (bytes), all offsets 16B-aligned
    char* ws = (char*)d_ws;
    float* bufA  = (float*)(ws + 0);            // N*256 f32 = 51.2 MB (XW)
    float* bufB  = (float*)(ws + 51200000);     // N*256 f32 (aggregation)
    float* bufC  = (float*)(ws + 102400000);    // N*256 f32 (layer input)
    float* dinv  = (float*)(ws + 153600000);    // N f32 (deg then dinv in place)
    float* normw = (float*)(ws + 153800000);    // E f32
    int*   src32 = (int*)  (ws + 157000000);    // E i32
    int*   dst32 = (int*)  (ws + 160200000);    // E i32

    const int T = 256;

    // --- edge prep ---
    init_deg_kernel<<<cdiv_i(N_NODES, T), T, 0, stream>>>(dinv);
    prep_edges_kernel<<<cdiv_i(N_EDGES, T), T, 0, stream>>>(ei, src32, dst32, dinv);
    dinv_kernel<<<cdiv_i(N_NODES, T), T, 0, stream>>>(dinv);
    norm_kernel<<<cdiv_i(N_EDGES, T), T, 0, stream>>>(src32, dst32, dinv, normw);

    const int mtiles = N_NODES / 16;            // 3125
    dim3 gblk(128);                             // 4 waves / block, 1 wave = 1 tile

    // --- Layer 1: 128 -> 256 ---
    gemm_wmma_f32<<<dim3(cdiv_i(mtiles, 4), 256 / 16), gblk, 0, stream>>>(x, W1, bufA, bufB, dinv, 128, 256);
    scatter_vec4_kernel<256><<<cdiv_i((long long)N_EDGES * 64, T), T, 0, stream>>>(bufA, src32, dst32, normw, bufB);
    bias_bn_relu_kernel<256><<<cdiv_i((long long)N_NODES * 256, T), T, 0, stream>>>(bufB, bufC, b1, g1, be1, m1, v1);

    // --- Layer 2: 256 -> 128 ---
    gemm_wmma_f32<<<dim3(cdiv_i(mtiles, 4), 128 / 16), gblk, 0, stream>>>(bufC, W2, bufA, bufB, dinv, 256, 128);
    scatter_vec4_kernel<128><<<cdiv_i((long long)N_EDGES * 32, T), T, 0, stream>>>(bufA, src32, dst32, normw, bufB);
    bias_bn_relu_kernel<128><<<cdiv_i((long long)N_NODES * 128, T), T, 0, stream>>>(bufB, bufC, b2, g2, be2, m2, v2);

    // --- Layer 3: 128 -> 64 ---
    gemm_wmma_f32<<<dim3(cdiv_i(mtiles, 4), 64 / 16), gblk, 0, stream>>>(bufC, W3, bufA, bufB, dinv, 128, 64);
    scatter_vec4_kernel<64><<<cdiv_i((long long)N_EDGES * 16, T), T, 0, stream>>>(bufA, src32, dst32, normw, bufB);
    bias_bn_relu_kernel<64><<<cdiv_i((long long)N_NODES * 64, T), T, 0, stream>>>(bufB, bufC, b3, g3, be3, m3, v3);

    // --- Layer 4: 64 -> 6 ---
    gemm4_kernel<<<cdiv_i(N_NODES, T), T, 0, stream>>>(bufC, W4, bufA, bufB, dinv);
    scatter_f6_kernel<<<cdiv_i(N_EDGES, T), T, 0, stream>>>(bufA, src32, dst32, normw, bufB);
    final_bias_kernel<<<cdiv_i((long long)N_NODES * 6, T), T, 0, stream>>>(bufB, b4, out);
}